// MLPTrajDecoder_64845416235028
// MI455X (gfx1250) — compile-verified
//
#include <hip/hip_runtime.h>

// MLPTrajDecoder for MI455X (gfx1250, wave32).
// GEMM1 (768x8 per pair-group) runs on V_WMMA_F32_16X16X4_F32 (K=8 via two
// chained WMMAs). Second layer is per-lane FMA + one cross-half shuffle.
// Cumsum kept in registers; backward pass served from LDS. Global
// traffic is b128 except 2-float Z/W1 loads; outputs use non-temporal stores (write-once streaming).

typedef __attribute__((ext_vector_type(2))) float v2f;
typedef __attribute__((ext_vector_type(4))) float v4f;
typedef __attribute__((ext_vector_type(8))) float v8f;

#define NB 256
#define NA 512
#define NT_FULL 49          // T
#define NIN 4
#define NEMB 16
#define NLAT 8
#define NTILE 48            // T-1 tiles of 16 embeddings
#define WAVES 4
#define NPAIR (NB * NA)     // 131072
#define NGROUP (NPAIR / 16) // 8192 groups of 16 pairs

__device__ __forceinline__ float elu1(float x) {
  return x > 0.0f ? x : (__expf(x) - 1.0f);
}

__global__ __launch_bounds__(WAVES * 32)
void traj_decoder_kernel(const float* __restrict__ Z,
                         const float* __restrict__ X,
                         const float* __restrict__ W1,
                         const float* __restrict__ b1,
                         const float* __restrict__ W2,
                         const float* __restrict__ b2,
                         float* __restrict__ out_fwd,
                         float* __restrict__ out_bwd)
{
  __shared__ __align__(16) float lds_cum[WAVES][NTILE][16][4];

  const int tid  = threadIdx.x;
  const int wave = tid >> 5;
  const int lane = tid & 31;
  const int p    = lane & 15;   // pair within group
  const int half = lane >> 4;   // 0: K/e low half, 1: high half
  const int kb   = half * 2;    // K base for this lane's A/B VGPRs (0 or 2)
  const int e0   = half * 8;    // embedding sub-range handled by this lane

  const long group = (long)blockIdx.x * WAVES + wave;
  const long pair  = group * 16 + p;

  // ---- B operand: Z tile (K=8 x N=16 pairs), invariant over t-loop ----
  // B layout (4x16, 32-bit): VGPR0 lane n<16 -> B[K=0,N=n], lane n>=16 -> B[K=2,N=n-16];
  //                          VGPR1 -> K=1 / K=3.
  const float* zp = Z + pair * NLAT;
  v2f bz0, bz1;
  bz0.x = zp[kb + 0]; bz0.y = zp[kb + 1];   // K = kb, kb+1       (WMMA #1)
  bz1.x = zp[kb + 4]; bz1.y = zp[kb + 5];   // K = kb+4, kb+5     (WMMA #2)

  // ---- W2 (4x16) per-lane embedding half, b2 broadcast ----
  float w2v[4][8];
#pragma unroll
  for (int i = 0; i < 4; ++i) {
    v4f lo = *(const v4f*)(W2 + i * NEMB + e0);
    v4f hi = *(const v4f*)(W2 + i * NEMB + e0 + 4);
#pragma unroll
    for (int r = 0; r < 4; ++r) { w2v[i][r] = lo[r]; w2v[i][4 + r] = hi[r]; }
  }
  v4f b2v = *(const v4f*)b2;

  // ---- trajectory endpoints for this lane's pair ----
  const float* xb = X + pair * (NT_FULL * NIN);
  v4f x0 = *(const v4f*)(xb);
  v4f xt = *(const v4f*)(xb + (NT_FULL - 1) * NIN);

  float* fwdp = out_fwd + pair * (NT_FULL * NIN);
  float* bwdp = out_bwd + pair * (NT_FULL * NIN);

  if (half == 0) {
    __builtin_nontemporal_store(x0, (v4f*)fwdp);       // forward[t=0] = X[:, :, 0]
  }

  v4f cum = {0.f, 0.f, 0.f, 0.f};

#pragma unroll 1
  for (int t = 0; t < NTILE; ++t) {
    // A operand: W1 tile rows m = t*16 .. t*16+15, K=8 split over 2 WMMAs.
    // A layout (16x4, 32-bit): lane n<16 VGPR0 -> A[M=n,K=0], VGPR1 -> K=1;
    //                          lane n>=16     -> K=2 / K=3.
    const float* w1p = W1 + (long)(t * 16 + p) * NLAT + kb;
    v2f a0, a1;
    a0.x = w1p[0]; a0.y = w1p[1];
    a1.x = w1p[4]; a1.y = w1p[5];

    v8f c = {};
    c = __builtin_amdgcn_wmma_f32_16x16x4_f32(false, a0, false, bz0,
                                              (short)0, c, false, false);
    c = __builtin_amdgcn_wmma_f32_16x16x4_f32(false, a1, false, bz1,
                                              (short)0, c, false, false);

    // C layout: lane n<16 VGPR r -> C[M=r, N=n]  => h[e0+r, pair p] in c[r].
    const float* b1p = b1 + t * NEMB + e0;
    v4f blo = *(const v4f*)(b1p);
    v4f bhi = *(const v4f*)(b1p + 4);
    float hv[8];
#pragma unroll
    for (int r = 0; r < 4; ++r) {
      hv[r]     = elu1(c[r]     + blo[r]);
      hv[4 + r] = elu1(c[4 + r] + bhi[r]);
    }

    // Second layer partials: part[i] = sum_r h[e0+r] * W2[i, e0+r]
    v4f part = {0.f, 0.f, 0.f, 0.f};
#pragma unroll
    for (int r = 0; r < 8; ++r) {
      part.x += hv[r] * w2v[0][r];
      part.y += hv[r] * w2v[1][r];
      part.z += hv[r] * w2v[2][r];
      part.w += hv[r] * w2v[3][r];
    }
    // Combine embedding halves: lane p <-> lane p+16 (wave32 xor-16 shuffle)
    v4f full;
    full.x = part.x + __shfl_xor(part.x, 16, 32);
    full.y = part.y + __shfl_xor(part.y, 16, 32);
    full.z = part.z + __shfl_xor(part.z, 16, 32);
    full.w = part.w + __shfl_xor(part.w, 16, 32);

    cum += full + b2v;                                  // running prefix sum

    if (half == 0) {                                    // forward[t+1] = X0 + cum
      v4f f = x0 + cum;
      __builtin_nontemporal_store(f, (v4f*)(fwdp + (t + 1) * NIN));
    } else {                                            // park cum[t] for pass 2
      *(v4f*)(&lds_cum[wave][t][p][0]) = cum;
    }
  }

  __syncthreads();

  // backward[0]   = XT - total
  // backward[t>0] = XT - total + cum[t-1]
  v4f base = xt - cum;                                  // cum == total here
#pragma unroll 1
  for (int j = 0; j < 25; ++j) {
    int t = 2 * j + half;                               // half 0: even t, half 1: odd t
    if (t >= NT_FULL) break;
    v4f cprev = {0.f, 0.f, 0.f, 0.f};
    if (t > 0) cprev = *(const v4f*)(&lds_cum[wave][t - 1][p][0]);
    v4f o = base + cprev;
    __builtin_nontemporal_store(o, (v4f*)(bwdp + t * NIN));
  }
}

extern "C" void kernel_launch(void* const* d_in, const int* in_sizes, int n_in,
                              void* d_out, int out_size, void* d_ws, size_t ws_size,
                              hipStream_t stream) {
  (void)in_sizes; (void)n_in; (void)out_size; (void)d_ws; (void)ws_size;
  const float* Z  = (const float*)d_in[0];
  const float* X  = (const float*)d_in[1];
  const float* W1 = (const float*)d_in[2];
  const float* b1 = (const float*)d_in[3];
  const float* W2 = (const float*)d_in[4];
  const float* b2 = (const float*)d_in[5];

  float* fwd = (float*)d_out;
  float* bwd = fwd + (size_t)NB * NA * NT_FULL * NIN;   // outputs concatenated

  dim3 block(WAVES * 32);
  dim3 grid(NGROUP / WAVES);                            // 2048 workgroups
  traj_decoder_kernel<<<grid, block, 0, stream>>>(Z, X, W1, b1, W2, b2, fwd, bwd);
}